// RelAttnEntPnaBlock_84559316123892
// MI455X (gfx1250) — compile-verified
//
#include <hip/hip_runtime.h>
#include <hip/hip_bf16.h>
#include <stdint.h>

typedef __attribute__((ext_vector_type(2))) float v2f;
typedef __attribute__((ext_vector_type(8))) float v8f;
typedef unsigned int u32x4 __attribute__((ext_vector_type(4)));
typedef int i32x4 __attribute__((ext_vector_type(4)));
typedef int i32x8 __attribute__((ext_vector_type(8)));

#define D_DIM 128

// ---- float atomic min/max via sign-aware integer atomics
__device__ inline void atomicMaxF(float* addr, float v) {
    if (v >= 0.f) atomicMax((int*)addr, __float_as_int(v));
    else          atomicMin((unsigned int*)addr, (unsigned int)__float_as_int(v));
}
__device__ inline void atomicMinF(float* addr, float v) {
    if (v >= 0.f) atomicMin((int*)addr, __float_as_int(v));
    else          atomicMax((unsigned int*)addr, (unsigned int)__float_as_int(v));
}

// ---- TDM: DMA one contiguous 64KB (16384 x f32) block from global into LDS.
__device__ inline void tdm_copy_16k_f32(const float* gsrc, float* lds_dst) {
    unsigned long long ga = (unsigned long long)(uintptr_t)gsrc;
    unsigned int       la = (unsigned int)(uintptr_t)lds_dst;   // LDS offset = low 32b
    u32x4 g0;
    g0.x = 1u;                                                  // count=1 (user mode)
    g0.y = la;                                                  // lds_addr  [63:32]
    g0.z = (unsigned int)ga;                                    // gaddr lo  [95:64]
    g0.w = ((unsigned int)(ga >> 32) & 0x01FFFFFFu) | 0x80000000u; // gaddr hi | type=2
    i32x8 g1;
    g1[0] = 0x20000;                 // workgroup_mask=0, data_size=2 (4 bytes)
    g1[1] = (int)(16384u << 16);     // tensor_dim0[15:0]
    g1[2] = (int)(1u << 16);         // tensor_dim1 = 1
    g1[3] = (int)(16384u << 16);     // tile_dim0 = 16384
    g1[4] = 0;
    g1[5] = 16384;                   // tensor_dim0_stride[31:0]
    g1[6] = 0;
    g1[7] = 0;
    i32x4 gz = {0, 0, 0, 0};
#if defined(__clang_major__) && __clang_major__ >= 23
    i32x8 gz8 = {0, 0, 0, 0, 0, 0, 0, 0};
    __builtin_amdgcn_tensor_load_to_lds(g0, g1, gz, gz, gz8, 0);
#else
    __builtin_amdgcn_tensor_load_to_lds(g0, g1, gz, gz, 0);
#endif
}

// ---- Async gather: one 16-byte chunk per lane, global -> LDS (ASYNCcnt path)
__device__ inline void async_g2l_b128(unsigned int lds_off, unsigned long long gaddr) {
    asm volatile("global_load_async_to_lds_b128 %0, %1, off"
                 :: "v"(lds_off), "v"(gaddr) : "memory");
}
__device__ inline void wait_asynccnt0() {
    asm volatile("s_wait_asynccnt 0" ::: "memory");
}

// ============================================================================
// Kernel 0a: init workspace aggregates
// ============================================================================
__global__ void init_ws_kernel(float* __restrict__ gsum, float* __restrict__ gmax,
                               float* __restrict__ gmin, float* __restrict__ gsq,
                               float* __restrict__ gcnt, long nD, int n) {
    long i = (long)blockIdx.x * blockDim.x + threadIdx.x;
    const float INF = __int_as_float(0x7f800000);
    if (i < nD) {
        gsum[i] = 0.f;
        gsq[i]  = 0.f;
        gmax[i] = -INF;
        gmin[i] =  INF;
    }
    if (i < n) gcnt[i] = 0.f;
}

// ============================================================================
// Kernel 0b: transpose weights so B-fragment K-pairs become contiguous b64s
// dst[c][r] = src[r][c]
// ============================================================================
__global__ void transpose_kernel(const float* __restrict__ src,
                                 float* __restrict__ dst, int rows, int cols) {
    int i = blockIdx.x * blockDim.x + threadIdx.x;
    if (i < rows * cols) {
        int r = i / cols, c = i % cols;
        dst[c * rows + r] = src[i];
    }
}

// ============================================================================
// Kernel 1: qn = q @ Wq   (fp32 WMMA 16x16x4, B from transposed Wq)
// ============================================================================
__global__ __launch_bounds__(256) void node_proj_kernel(
    const float* __restrict__ q, const float* __restrict__ Wqt,
    float* __restrict__ qn, int n) {
    __shared__ float qs[16][D_DIM];
    const int tid  = threadIdx.x;
    const int row0 = blockIdx.x * 16;

    for (int i = tid; i < 16 * D_DIM; i += 256) {
        int r = i >> 7, c = i & 127;
        qs[r][c] = (row0 + r < n) ? q[(long)(row0 + r) * D_DIM + c] : 0.f;
    }
    __syncthreads();

    const int wv   = tid >> 5;
    const int lane = tid & 31;
    const int M    = lane & 15;
    const int half = lane >> 4;
    const int koff = half * 2;
    const int colb = wv * 16;

    const float* bcol = Wqt + (long)(colb + M) * D_DIM;  // column (colb+M) of Wq
    v8f c = {};
    for (int k0 = 0; k0 < D_DIM; k0 += 4) {
        v2f a = *(const v2f*)(&qs[M][k0 + koff]);
        v2f b = *(const v2f*)(bcol + k0 + koff);
        c = __builtin_amdgcn_wmma_f32_16x16x4_f32(false, a, false, b, (short)0, c,
                                                  false, false);
    }
    for (int r = 0; r < 8; ++r) {
        int row = row0 + r + half * 8;
        if (row < n) qn[(long)row * D_DIM + colb + M] = c[r];
    }
}

// ============================================================================
// Kernel 2: edge pipeline.
//  - async-gather E[src] / R_emb[rel] / qn[dst] rows into LDS (ASYNCcnt)
//  - TDM-stage transposed Wk / Wv (64KB each) into LDS        (TENSORcnt)
//  - msg = E*R in place, two WMMA chains, sigmoid attn per head,
//    PNA scatter with float atomics.
// block = 128 threads (4 waves); each wave owns 16 edges.
// dynamic LDS: er/msg 32K | rr 32K | qe 32K | Wkt 64K | Wvt 64K = 224KB
// ============================================================================
__global__ __launch_bounds__(128) void edge_kernel(
    const float* __restrict__ Emb, const float* __restrict__ R,
    const float* __restrict__ Wkt, const float* __restrict__ Wvt,
    const float* __restrict__ qn,
    const int* __restrict__ esrc, const int* __restrict__ erel,
    const int* __restrict__ edst, const float* __restrict__ edeg,
    float* __restrict__ gsum, float* __restrict__ gmax,
    float* __restrict__ gmin, float* __restrict__ gsq,
    float* __restrict__ gcnt, int nE) {
    extern __shared__ __align__(16) char smem_raw[];
    float* er  = (float*)smem_raw;            // [4][16][128]  (becomes msg in place)
    float* rr  = er  + 4 * 16 * D_DIM;        // [4][16][128]
    float* qe  = rr  + 4 * 16 * D_DIM;        // [4][16][128]  gathered qn[dst]
    float* wks = qe  + 4 * 16 * D_DIM;        // [128][128]  (transposed Wk)
    float* wvs = wks + D_DIM * D_DIM;         // [128][128]  (transposed Wv)

    __shared__ int   src_s[4][16], rel_s[4][16], dst_s[4][16];
    __shared__ float deg_s[4][16];

    const int tid  = threadIdx.x;
    const int wv   = tid >> 5;
    const int lane = tid & 31;
    const int M    = lane & 15;
    const int half = lane >> 4;
    const int koff = half * 2;
    const long e0  = (long)blockIdx.x * 64 + (long)wv * 16;

    if (lane < 16) {
        long e = e0 + lane;
        int s = 0, r = 0, d = -1; float dg = 0.f;
        if (e < nE) { s = esrc[e]; r = erel[e]; d = edst[e]; dg = edeg[e]; }
        src_s[wv][lane] = s; rel_s[wv][lane] = r;
        dst_s[wv][lane] = d; deg_s[wv][lane] = dg;
        // speculative prefetch of next block's edge meta
        __builtin_prefetch(esrc + e + 64, 0, 0);
        __builtin_prefetch(edst + e + 64, 0, 0);
    }
    __syncthreads();

    // Async gather: 32 lanes x 16B == one 512B row per issue.
    for (int m = 0; m < 16; ++m) {
        const long sb = (long)src_s[wv][m] * D_DIM;
        const long rb = (long)rel_s[wv][m] * D_DIM;
        const int  dm = dst_s[wv][m];
        const long db = (long)(dm < 0 ? 0 : dm) * D_DIM;
        const int  rowoff = (wv * 16 + m) * D_DIM + lane * 4;
        async_g2l_b128((unsigned int)(uintptr_t)(er + rowoff),
                       (unsigned long long)(uintptr_t)(Emb + sb + lane * 4));
        async_g2l_b128((unsigned int)(uintptr_t)(rr + rowoff),
                       (unsigned long long)(uintptr_t)(R + rb + lane * 4));
        async_g2l_b128((unsigned int)(uintptr_t)(qe + rowoff),
                       (unsigned long long)(uintptr_t)(qn + db + lane * 4));
    }
    // TDM: wave 0 DMAs both (transposed) weight matrices into LDS.
    if (wv == 0) {
        tdm_copy_16k_f32(Wkt, wks);
        tdm_copy_16k_f32(Wvt, wvs);
        __builtin_amdgcn_s_wait_tensorcnt(0);
    }
    wait_asynccnt0();
    __syncthreads();

    // msg = E[src]*R[rel], in place over the E rows.
    for (int m = 0; m < 16; ++m) {
        const bool ok = dst_s[wv][m] >= 0;
        float* row_e = er + ((wv * 16 + m) * D_DIM);
        float* row_r = rr + ((wv * 16 + m) * D_DIM);
        for (int c = lane; c < D_DIM; c += 32)
            row_e[c] = ok ? row_e[c] * row_r[c] : 0.f;
    }
    __syncthreads();

    // segment counts (one lane per edge)
    if (lane < 16) {
        int d = dst_s[wv][lane];
        if (d >= 0) atomicAdd(&gcnt[d], 1.0f);
    }

    const float* msg  = er + (wv * 16) * D_DIM;  // this wave's 16x128 msg tile
    const float* qet  = qe + (wv * 16) * D_DIM;  // this wave's 16x128 qe tile
    const float* bkc  = wks + (long)M * D_DIM;   // col base updated per nb below
    const float* bvc  = wvs + (long)M * D_DIM;

    // 8 column blocks == 8 heads (HD == 16)
    for (int nb = 0; nb < 8; ++nb) {
        const int colb = nb * 16;
        const float* bk = bkc + (long)colb * D_DIM;  // column (colb+M) of Wk
        const float* bv = bvc + (long)colb * D_DIM;
        v8f ck = {};
        v8f cv = {};
        for (int k0 = 0; k0 < D_DIM; k0 += 4) {
            v2f a   = *(const v2f*)(msg + M * D_DIM + k0 + koff);
            v2f bkf = *(const v2f*)(bk + k0 + koff);
            v2f bvf = *(const v2f*)(bv + k0 + koff);
            ck = __builtin_amdgcn_wmma_f32_16x16x4_f32(false, a, false, bkf, (short)0,
                                                       ck, false, false);
            cv = __builtin_amdgcn_wmma_f32_16x16x4_f32(false, a, false, bvf, (short)0,
                                                       cv, false, false);
        }
        for (int r = 0; r < 8; ++r) {
            const int row = r + half * 8;
            const int d   = dst_s[wv][row];
            float qv = qet[row * D_DIM + colb + M];
            float p  = qv * ck[r];
            p += __shfl_xor(p, 1);
            p += __shfl_xor(p, 2);
            p += __shfl_xor(p, 4);
            p += __shfl_xor(p, 8);                 // 16-lane head dot-product
            const float alpha = p * 0.25f;         // 1/sqrt(HD)
            const float w = 1.f / (1.f + __expf(-alpha));
            const float x = w * deg_s[wv][row] * cv[r];
            if (d >= 0) {
                const long off = (long)d * D_DIM + colb + M;
                atomicAdd(&gsum[off], x);
                atomicMaxF(&gmax[off], x);
                atomicMinF(&gmin[off], x);
                atomicAdd(&gsq[off], x * x);
            }
        }
    }
}

// ============================================================================
// Kernel 3: PNA combine -> agg[16x512] in LDS -> WMMA agg@Wo (K=512, B from
//           transposed Wo) -> residual + LayerNorm -> d_out
// ============================================================================
__global__ __launch_bounds__(256) void finalize_kernel(
    const float* __restrict__ q,    const float* __restrict__ Wot,
    const float* __restrict__ ln_g, const float* __restrict__ ln_b,
    const float* __restrict__ gsum, const float* __restrict__ gmax,
    const float* __restrict__ gmin, const float* __restrict__ gsq,
    const float* __restrict__ gcnt, float* __restrict__ out, int n) {
    __shared__ float agg_s[16][4 * D_DIM];
    __shared__ float out_s[16][D_DIM];

    const int tid  = threadIdx.x;
    const int row0 = blockIdx.x * 16;

    for (int i = tid; i < 16 * D_DIM; i += 256) {
        int r = i >> 7, c = i & 127;
        int node = row0 + r;
        float s = 0.f, sq = 0.f, mx = 0.f, mn = 0.f, cct = 0.f;
        if (node < n) {
            long off = (long)node * D_DIM + c;
            s = gsum[off]; sq = gsq[off]; mx = gmax[off]; mn = gmin[off];
            cct = gcnt[node];
        }
        const bool  ne   = cct > 0.f;
        const float mean = ne ? s / fmaxf(cct, 1.f) : 0.f;
        const float sd   = sqrtf(fmaxf(sq - s * s, 1e-8f));
        agg_s[r][c]             = mean;
        agg_s[r][D_DIM + c]     = ne ? mx : 0.f;
        agg_s[r][2 * D_DIM + c] = ne ? mn : 0.f;
        agg_s[r][3 * D_DIM + c] = sd;
    }
    __syncthreads();

    const int wv   = tid >> 5;
    const int lane = tid & 31;
    const int M    = lane & 15;
    const int half = lane >> 4;
    const int koff = half * 2;
    const int colb = wv * 16;

    const float* bcol = Wot + (long)(colb + M) * (4 * D_DIM);  // column of Wo, K=512
    v8f c = {};
    for (int k0 = 0; k0 < 4 * D_DIM; k0 += 4) {
        v2f a = *(const v2f*)(&agg_s[M][k0 + koff]);
        v2f b = *(const v2f*)(bcol + k0 + koff);
        c = __builtin_amdgcn_wmma_f32_16x16x4_f32(false, a, false, b, (short)0, c,
                                                  false, false);
    }
    for (int r = 0; r < 8; ++r) {
        int row = r + half * 8;
        int node = row0 + row;
        float resid = (node < n) ? q[(long)node * D_DIM + colb + M] : 0.f;
        out_s[row][colb + M] = c[r] + resid;
    }
    __syncthreads();

    const int row = tid >> 4;
    const int sub = tid & 15;
    float sum = 0.f;
    for (int cc = sub; cc < D_DIM; cc += 16) sum += out_s[row][cc];
    sum += __shfl_xor(sum, 1); sum += __shfl_xor(sum, 2);
    sum += __shfl_xor(sum, 4); sum += __shfl_xor(sum, 8);
    const float mu = sum * (1.f / D_DIM);
    float vs = 0.f;
    for (int cc = sub; cc < D_DIM; cc += 16) {
        float dlt = out_s[row][cc] - mu;
        vs += dlt * dlt;
    }
    vs += __shfl_xor(vs, 1); vs += __shfl_xor(vs, 2);
    vs += __shfl_xor(vs, 4); vs += __shfl_xor(vs, 8);
    const float inv = rsqrtf(vs * (1.f / D_DIM) + 1e-5f);
    const int node = row0 + row;
    if (node < n) {
        for (int cc = sub; cc < D_DIM; cc += 16)
            out[(long)node * D_DIM + cc] =
                (out_s[row][cc] - mu) * inv * ln_g[cc] + ln_b[cc];
    }
}

// ============================================================================
extern "C" void kernel_launch(void* const* d_in, const int* in_sizes, int n_in,
                              void* d_out, int out_size, void* d_ws, size_t ws_size,
                              hipStream_t stream) {
    const float* q    = (const float*)d_in[0];
    const float* Emb  = (const float*)d_in[1];
    const float* R    = (const float*)d_in[2];
    const float* Wq   = (const float*)d_in[3];
    const float* Wk   = (const float*)d_in[4];
    const float* Wv   = (const float*)d_in[5];
    const float* Wo   = (const float*)d_in[6];
    const float* ln_g = (const float*)d_in[7];
    const float* ln_b = (const float*)d_in[8];
    const float* deg  = (const float*)d_in[9];
    const int*   esrc = (const int*)d_in[10];
    const int*   erel = (const int*)d_in[11];
    const int*   edst = (const int*)d_in[12];

    const int  n  = in_sizes[0] / D_DIM;
    const int  nE = in_sizes[10];
    const long nD = (long)n * D_DIM;

    float* ws   = (float*)d_ws;
    float* qn   = ws;
    float* gsum = ws + nD;
    float* gmax = ws + 2 * nD;
    float* gmin = ws + 3 * nD;
    float* gsq  = ws + 4 * nD;
    float* gcnt = ws + 5 * nD;             // [n]
    float* wxt  = gcnt + n;                // transposed weights
    float* Wqt  = wxt;                     // [128*128]
    float* Wkt  = Wqt + D_DIM * D_DIM;     // [128*128]
    float* Wvt  = Wkt + D_DIM * D_DIM;     // [128*128]
    float* Wot  = Wvt + D_DIM * D_DIM;     // [128*512]
    float* out  = (float*)d_out;

    const int edgeSmem = (4 * 16 * D_DIM * 3 + 2 * D_DIM * D_DIM) * (int)sizeof(float);
    (void)hipFuncSetAttribute((const void*)edge_kernel,
                              hipFuncAttributeMaxDynamicSharedMemorySize, edgeSmem);

    init_ws_kernel<<<(int)((nD + 255) / 256), 256, 0, stream>>>(
        gsum, gmax, gmin, gsq, gcnt, nD, n);
    transpose_kernel<<<(D_DIM * D_DIM + 255) / 256, 256, 0, stream>>>(
        Wq, Wqt, D_DIM, D_DIM);
    transpose_kernel<<<(D_DIM * D_DIM + 255) / 256, 256, 0, stream>>>(
        Wk, Wkt, D_DIM, D_DIM);
    transpose_kernel<<<(D_DIM * D_DIM + 255) / 256, 256, 0, stream>>>(
        Wv, Wvt, D_DIM, D_DIM);
    transpose_kernel<<<(4 * D_DIM * D_DIM + 255) / 256, 256, 0, stream>>>(
        Wo, Wot, 4 * D_DIM, D_DIM);
    node_proj_kernel<<<(n + 15) / 16, 256, 0, stream>>>(q, Wqt, qn, n);
    edge_kernel<<<(nE + 63) / 64, 128, edgeSmem, stream>>>(
        Emb, R, Wkt, Wvt, qn, esrc, erel, edst, deg,
        gsum, gmax, gmin, gsq, gcnt, nE);
    finalize_kernel<<<(n + 15) / 16, 256, 0, stream>>>(
        q, Wot, ln_g, ln_b, gsum, gmax, gmin, gsq, gcnt, out, n);
}